// VQVAE_51204600103559
// MI455X (gfx1250) — compile-verified
//
#include <hip/hip_runtime.h>

typedef float v2f __attribute__((ext_vector_type(2)));
typedef float v8f __attribute__((ext_vector_type(8)));

#define N_PIX   32768
#define K_CODES 8192
#define HW      4096
#define CHW     16384
#define TILES   (K_CODES / 16)      // 512
#define THREADS 256
#define WAVES   8                   // 256 threads / wave32
#define PIX_PER_BLOCK (WAVES * 16)  // 128
#define NBLOCKS (N_PIX / PIX_PER_BLOCK) // 256

// Dynamic LDS layout (bytes) — exactly 160 KB so TWO workgroups fit in the
// 320 KB WGP LDS (16 waves / WGP for latency hiding):
//   [0,       131072)  codebook, 8192 codes x 4 floats (float4 aligned)
//   [131072,  163840)  norms, 8192 floats  -- only live during the tile loop
// After the post-loop barrier the norm region is dead, so we overlay:
//   [131072,  131584)  per-wave winner indices, WAVES*16 ints (wave-local)
//   [131584,  132608)  loss reduction scratch, 256 floats
#define LDS_CB_OFF   0
#define LDS_NORM_OFF 131072
#define LDS_IDX_OFF  131072
#define LDS_RED_OFF  131584
#define LDS_BYTES    163840

extern "C" __global__ void vq_main(const float* __restrict__ z,
                                   const float* __restrict__ cb,
                                   float* __restrict__ out,
                                   float* __restrict__ partial) {
    extern __shared__ char smem[];
    float* lds_cb   = (float*)(smem + LDS_CB_OFF);
    float* lds_norm = (float*)(smem + LDS_NORM_OFF);
    int*   lds_idx  = (int*)  (smem + LDS_IDX_OFF);   // overlays norms (post-loop)
    float* lds_red  = (float*)(smem + LDS_RED_OFF);   // overlays norms (post-loop)

    const int tid = threadIdx.x;

    // ---- Stage codebook + norms into LDS
    for (int k = tid; k < K_CODES; k += THREADS) {
        const float4 e = ((const float4*)cb)[k];       // global_load_b128
        ((float4*)lds_cb)[k] = e;                      // ds_store_b128
        lds_norm[k] = e.x*e.x + e.y*e.y + e.z*e.z + e.w*e.w;
    }
    __syncthreads();

    const int wave = tid >> 5;
    const int lane = tid & 31;
    const int half = lane >> 4;     // 0: K=0,1  1: K=2,3 (A layout)
    const int l16  = lane & 15;
    const int p0   = blockIdx.x * PIX_PER_BLOCK + wave * 16;

    // ---- Build A = -2 * z tile, 16x4 f32 (2 VGPRs per lane)
    v2f a;
    {
        const int n  = p0 + l16;        // pixel (M = l16)
        const int bb = n >> 12;         // n / (H*W)
        const int hw = n & (HW - 1);
        const int c0 = half * 2;
        const float z0 = z[bb * CHW + (c0    ) * HW + hw];
        const float z1 = z[bb * CHW + (c0 + 1) * HW + hw];
        a.x = -2.0f * z0;               // VGPR0: K = c0
        a.y = -2.0f * z1;               // VGPR1: K = c0+1
    }

    // ---- Sweep all codes: D = A*B + C  with  B = e columns, C = |e|^2
    float minval[8];
    int   minidx[8];
#pragma unroll
    for (int j = 0; j < 8; ++j) { minval[j] = 3.402823466e38f; minidx[j] = 0; }

    // B layout: lanes 0-15 -> (K=0,1), lanes 16-31 -> (K=2,3) for code N = l16
    const float* bptr = lds_cb + (l16 * 4 + half * 2);   // 8B aligned -> ds_load_b64
    const float* nptr = lds_norm + l16;

#pragma unroll 4
    for (int t = 0; t < TILES; ++t) {
        const v2f   b   = *(const v2f*)bptr;
        const float nrm = *nptr;
        v8f c = { nrm, nrm, nrm, nrm, nrm, nrm, nrm, nrm };
        // score = |e|^2 - 2 z.e   (|z|^2 constant dropped for argmin)
        v8f d = __builtin_amdgcn_wmma_f32_16x16x4_f32(
            /*neg_a=*/false, a, /*neg_b=*/false, b,
            /*c_mod=*/(short)0, c, /*reuse_a=*/false, /*reuse_b=*/false);
        const int k = (t << 4) + l16;
#pragma unroll
        for (int j = 0; j < 8; ++j) {
            if (d[j] < minval[j]) { minval[j] = d[j]; minidx[j] = k; }
        }
        bptr += 64;   // 16 codes * 4 floats
        nptr += 16;
    }

    // Barrier A: all waves finished reading the norm region -> overlay is safe.
    __syncthreads();

    // ---- Cross-lane argmin within each 16-lane column group (masks < 16 stay in group)
#pragma unroll
    for (int j = 0; j < 8; ++j) {
        float v  = minval[j];
        int   ix = minidx[j];
#pragma unroll
        for (int m = 8; m >= 1; m >>= 1) {
            const float ov = __shfl_xor(v, m, 32);
            const int   oi = __shfl_xor(ix, m, 32);
            if (ov < v || (ov == v && oi < ix)) { v = ov; ix = oi; }
        }
        minidx[j] = ix;   // pixel m = j + 8*half, result in all 16 lanes of group
    }

    // publish winners (wave-local exchange; LDS ops are in-order within a wave)
    if (l16 == 0) {
#pragma unroll
        for (int j = 0; j < 8; ++j) lds_idx[wave * 16 + half * 8 + j] = minidx[j];
    }
    __syncthreads();

    // ---- Gather z_q, write output, accumulate loss partial
    float lsum = 0.0f;
    if (l16 < 8) {                      // 16 handling lanes per wave
        const int m  = half * 8 + l16;
        const int ix = lds_idx[wave * 16 + m];
        const int n  = p0 + m;
        const int bb = n >> 12;
        const int hw = n & (HW - 1);
        const float4 e = ((const float4*)lds_cb)[ix];
        const float zq[4] = { e.x, e.y, e.z, e.w };
#pragma unroll
        for (int c = 0; c < 4; ++c) {
            const int addr = bb * CHW + c * HW + hw;
            const float diff = zq[c] - z[addr];
            lsum += diff * diff;
            out[addr] = zq[c];          // z_q_st == z_q numerically
        }
    }

    // ---- Deterministic fixed-order block reduction of loss partials
    lds_red[tid] = lsum;
    __syncthreads();
    for (int s = THREADS / 2; s > 0; s >>= 1) {
        if (tid < s) lds_red[tid] += lds_red[tid + s];
        __syncthreads();
    }
    if (tid == 0) partial[blockIdx.x] = lds_red[0];
}

extern "C" __global__ void vq_loss(const float* __restrict__ partial,
                                   float* __restrict__ out_loss) {
    __shared__ float red[NBLOCKS];
    const int tid = threadIdx.x;
    red[tid] = partial[tid];
    __syncthreads();
    for (int s = NBLOCKS / 2; s > 0; s >>= 1) {
        if (tid < s) red[tid] += red[tid + s];
        __syncthreads();
    }
    // loss = (BETA + 1) * mean((z_q - z)^2) over B*C*H*W elements
    if (tid == 0) out_loss[0] = 1.25f * red[0] / 131072.0f;
}

extern "C" void kernel_launch(void* const* d_in, const int* in_sizes, int n_in,
                              void* d_out, int out_size, void* d_ws, size_t ws_size,
                              hipStream_t stream) {
    const float* z  = (const float*)d_in[0];   // [8,4,64,64] f32
    const float* cb = (const float*)d_in[1];   // [8192,4]    f32
    float* out      = (float*)d_out;           // 131072 z_q + 1 loss
    float* partial  = (float*)d_ws;            // 256 block partials

    (void)in_sizes; (void)n_in; (void)out_size; (void)ws_size;

    // Exactly 160 KB dynamic LDS -> two workgroups per 320 KB WGP.
    hipFuncSetAttribute(reinterpret_cast<const void*>(vq_main),
                        hipFuncAttributeMaxDynamicSharedMemorySize, LDS_BYTES);

    vq_main<<<NBLOCKS, THREADS, LDS_BYTES, stream>>>(z, cb, out, partial);
    vq_loss<<<1, NBLOCKS, 0, stream>>>(partial, out + N_PIX * 4);
}